// Unfold3d_51187420233901
// MI455X (gfx1250) — compile-verified
//
#include <hip/hip_runtime.h>

typedef float v4f __attribute__((ext_vector_type(4)));
typedef int   v4i __attribute__((ext_vector_type(4)));

// Problem constants (from the reference): x[4,32,16,48,48], K=3, PAD=1, STR=1, DIL=1
#define B_  4
#define C_  32
#define D_  16
#define H_  48
#define W_  48
#define HW_ (H_ * W_)          // 2304 floats per plane
#define OUTPLANE_ (D_ * HW_)   // 36864 floats per (b, c*27+p) row

#if defined(__has_builtin)
#if __has_builtin(__builtin_amdgcn_global_load_async_to_lds_b128)
#define UNFOLD_HAS_ASYNC_LDS 1
#endif
#endif

typedef __attribute__((address_space(1))) v4i* glb_v4i_ptr;
typedef __attribute__((address_space(3))) v4i* lds_v4i_ptr;

__global__ __launch_bounds__(256)
void Unfold3d_51187420233901_kernel(const float* __restrict__ x,
                                    float* __restrict__ out) {
    // One workgroup per (b, c, od): stage the 3 needed depth planes in LDS,
    // then emit all 27 shifted copies with contiguous b128 NT stores.
    __shared__ __align__(16) float lds[3 * HW_];   // 27,648 bytes, 16B-aligned for b128

    const int wg  = blockIdx.x;      // wg = (b*C + c)*D + od
    const int od  = wg % D_;
    const int bc  = wg / D_;         // b*C + c
    const int tid = threadIdx.x;

    // ---- Stage planes id = od-1, od, od+1 into LDS (zero-fill when OOB) ----
    for (int kd = 0; kd < 3; ++kd) {
        const int id = od + kd - 1;
        v4f* ldst = ((v4f*)lds) + kd * (HW_ / 4);
        if ((unsigned)id < (unsigned)D_) {
            const v4f* src = (const v4f*)(x + (size_t)(bc * D_ + id) * HW_);
            for (int j = tid; j < HW_ / 4; j += 256) {   // 576 v4f; wave-aligned tail
#if defined(UNFOLD_HAS_ASYNC_LDS)
                __builtin_amdgcn_global_load_async_to_lds_b128(
                    (glb_v4i_ptr)(src + j),
                    (lds_v4i_ptr)(ldst + j),
                    0, 0);
#else
                ldst[j] = src[j];
#endif
            }
        } else {
            const v4f z = (v4f)(0.0f);
            for (int j = tid; j < HW_ / 4; j += 256) ldst[j] = z;
        }
    }
#if defined(UNFOLD_HAS_ASYNC_LDS)
#if __has_builtin(__builtin_amdgcn_s_wait_asynccnt)
    __builtin_amdgcn_s_wait_asynccnt(0);
#else
    asm volatile("s_wait_asynccnt 0" ::: "memory");
#endif
#endif
    __syncthreads();

    // ---- Emit 27 shifted copies ----
    // out element: out[(bc*27 + p)*OUTPLANE + od*HW + oh*48 + ow]
    float* outbase = out + (size_t)bc * 27 * OUTPLANE_ + (size_t)od * HW_;

    for (int p = 0; p < 27; ++p) {
        const int kd = p / 9;
        const int kh = (p / 3) % 3;
        const int kw = p % 3;                 // uniform across the inner loop
        const float* plane = lds + kd * HW_;
        float* outp = outbase + (size_t)p * OUTPLANE_;

        // 576 float4 per patch plane slice (48 rows x 12 vec4); contiguous per wave
        for (int i = tid; i < 576; i += 256) {
            const int oh = i / 12;
            const int w4 = i - oh * 12;
            const int ih = oh + kh - 1;
            v4f v = (v4f)(0.0f);
            if ((unsigned)ih < (unsigned)H_) {
                const float* row = plane + ih * W_;
                const v4f a = *(const v4f*)(row + w4 * 4);   // aligned ds_load_b128
                if (kw == 1) {
                    v = a;
                } else if (kw == 0) {
                    // elements row[w4*4-1 .. w4*4+2]
                    const float prev = (w4 > 0) ? row[w4 * 4 - 1] : 0.0f;
                    v = (v4f){prev, a.x, a.y, a.z};
                } else {
                    // elements row[w4*4+1 .. w4*4+4]
                    const float nxt = (w4 < 11) ? row[w4 * 4 + 4] : 0.0f;
                    v = (v4f){a.y, a.z, a.w, nxt};
                }
            }
            // 510 MB write-once stream: bypass/NT so it doesn't thrash L2
            __builtin_nontemporal_store(v, (v4f*)(outp + oh * W_ + w4 * 4));
        }
    }
}

extern "C" void kernel_launch(void* const* d_in, const int* in_sizes, int n_in,
                              void* d_out, int out_size, void* d_ws, size_t ws_size,
                              hipStream_t stream) {
    (void)in_sizes; (void)n_in; (void)d_ws; (void)ws_size; (void)out_size;
    const float* x = (const float*)d_in[0];
    float* out = (float*)d_out;
    // One workgroup per (b, c, od) slab
    dim3 grid(B_ * C_ * D_);
    dim3 block(256);
    Unfold3d_51187420233901_kernel<<<grid, block, 0, stream>>>(x, out);
}